// FrequencyAwareResampler_14886356648177
// MI455X (gfx1250) — compile-verified
//
#include <hip/hip_runtime.h>
#include <stdint.h>

// Problem constants (from reference): B=4, C=64, H=W=160, G=4, S=2.
#define Bc   4
#define Cc   64
#define Hc   160
#define Wc   160
#define Gc   4
#define CPG  16          // channels per group = C/G
#define TS   16          // low-res tile edge (16x16 pixels per block)
#define HFT  18          // max high-freq halo tile edge (TS + 2)
#define sHc  320
#define sWc  320

typedef unsigned int        u32;
typedef unsigned long long  u64;
typedef u32 v4u __attribute__((ext_vector_type(4)));
typedef int v4i __attribute__((ext_vector_type(4)));
typedef int v8i __attribute__((ext_vector_type(8)));

// Build a Tensor DMA Descriptor (D#) per CDNA5 ISA ch.8 and issue
// TENSOR_LOAD_TO_LDS. All sizes/strides in 4-byte elements (data_size=4B).
// Dim order: x (contiguous, tile_dim0), y (stride sy, tile_dim1),
// z (stride sz, tile_dim2), zz (stride szz, tile_dim3).
__device__ __forceinline__ void tdm_load_tile(
    u32 lds_byte, const void* gaddr,
    u32 tw, u32 th, u32 tz, u32 tzz,
    u64 sy, u64 sz, u64 szz,
    u32 td0, u32 td1, u32 td2, u32 td3)
{
  u64 ga = (u64)(uintptr_t)gaddr;
  v4u g0;
  g0.x = 1u;                                     // count=1, user descriptor
  g0.y = lds_byte;                               // LDS byte address
  g0.z = (u32)ga;                                // global_addr[31:0]
  g0.w = (u32)((ga >> 32) & 0x01FFFFFFu)         // global_addr[56:32]
         | (2u << 30);                           // type=2 ("image")
  v8i g1;
  g1[0] = (int)(2u << 16);                       // data_size=2 -> 4 bytes; no pad/iterate/multicast
  g1[1] = (int)((td0 & 0xFFFFu) << 16);          // tensor_dim0[15:0]
  g1[2] = (int)(((td0 >> 16) & 0xFFFFu) | ((td1 & 0xFFFFu) << 16));
  g1[3] = (int)(((td1 >> 16) & 0xFFFFu) | ((tw  & 0xFFFFu) << 16));   // tile_dim0
  g1[4] = (int)((th & 0xFFFFu) | ((tz & 0xFFFFu) << 16));             // tile_dim1, tile_dim2
  g1[5] = (int)(u32)(sy & 0xFFFFFFFFu);                               // dim0_stride[31:0]
  g1[6] = (int)(((u32)(sy >> 32) & 0xFFFFu) | (((u32)sz & 0xFFFFu) << 16));
  g1[7] = (int)(u32)((sz >> 16) & 0xFFFFFFFFu);                       // dim1_stride[47:16]
  v4i g2;
  g2[0] = (int)td2;                              // tensor_dim2
  g2[1] = (int)td3;                              // tensor_dim3 (iterate disabled)
  g2[2] = (int)(u32)(szz & 0xFFFFFFFFu);         // tensor_dim2_stride[31:0]
  g2[3] = (int)(((u32)(szz >> 32) & 0xFFFFu) | ((tzz & 0xFFFFu) << 16)); // tile_dim3
  v4i g3 = {0, 0, 0, 0};                         // no dim4
  v8i gx = {0, 0, 0, 0, 0, 0, 0, 0};             // unused extension group (6-arg form)
  __builtin_amdgcn_tensor_load_to_lds(g0, g1, g2, g3, gx, 0);
}

__global__ __launch_bounds__(256) void freq_resampler_kernel(
    const float* __restrict__ feature,
    const float* __restrict__ offset,
    const float* __restrict__ high_freq,
    float* __restrict__ out)
{
  __shared__ float lds_off[2 * 4 * TS * TS];       // [d][subpix k][y][x], 8 KB
  __shared__ float lds_hf[2][3 * HFT * HFT];       // double-buffered [band][y][x]

  const int b    = blockIdx.z;
  const int g    = blockIdx.y;
  const int tile = blockIdx.x;
  const int tx0  = (tile % (Wc / TS)) * TS;
  const int ty0  = (tile / (Wc / TS)) * TS;

  const int t  = threadIdx.x;
  const int lx = t & (TS - 1);
  const int ly = t >> 4;
  const int w  = tx0 + lx;
  const int h  = ty0 + ly;

  // High-freq halo tile, clamped to the tensor (border clamp == resize edge)
  const int r0 = max(ty0 - 1, 0);
  const int r1 = min(ty0 + TS, Hc - 1);
  const int c0 = max(tx0 - 1, 0);
  const int c1 = min(tx0 + TS, Wc - 1);
  const int rh = r1 - r0 + 1;                      // 17 or 18 (uniform)
  const int rw = c1 - c0 + 1;

  const bool wave0 = (t >> 5) == 0;

  const u32 lds_off_addr = (u32)(uintptr_t)&lds_off[0];
  const u32 lds_hf_addr0 = (u32)(uintptr_t)&lds_hf[0][0];
  const u32 lds_hf_addr1 = (u32)(uintptr_t)&lds_hf[1][0];

  // Offsets: channels (d*G + g)*4 + k, k = sy*2+sx.  z = k (4, stride H*W),
  // zz = d (2, stride 16*H*W).  Tile starts at channel 4g, pixel (ty0, tx0).
  const float* off_g = offset +
      (((size_t)b * (2 * Gc * 4) + 4 * g) * Hc + ty0) * (size_t)Wc + tx0;

  if (wave0) {
    tdm_load_tile(lds_off_addr, off_g,
                  TS, TS, 4, 2,
                  (u64)Wc, (u64)Hc * Wc, (u64)16 * Hc * Wc,
                  Wc, Hc, 4, 2);
    const float* hf_g = high_freq +
        (((size_t)(b * Cc + g * CPG + 0) * 3) * Hc + r0) * (size_t)Wc + c0;
    tdm_load_tile(lds_hf_addr0, hf_g,
                  (u32)rw, (u32)rh, 3, 0,
                  (u64)Wc, (u64)Hc * Wc, 0,
                  Wc, Hc, 3, 0);
    __builtin_amdgcn_s_wait_tensorcnt(0);
  }
  __syncthreads();

  // ---- per-thread sampling state (shared across the 16 channels) ----
  int   ix0[2][2], ix1[2][2], iy0[2][2], iy1[2][2];
  float fx[2][2], fy[2][2];
#pragma unroll
  for (int sy = 0; sy < 2; ++sy) {
#pragma unroll
    for (int sx = 0; sx < 2; ++sx) {
      const int k = sy * 2 + sx;
      const float offx = lds_off[(k * TS + ly) * TS + lx];
      const float offy = lds_off[((4 + k) * TS + ly) * TS + lx];
      // Replicate reference math: normalize -> unnormalize -> clamp
      float xn = 2.0f * ((float)w + 0.5f + offx) * (1.0f / (float)Wc) - 1.0f;
      float x  = (xn + 1.0f) * (0.5f * (float)Wc) - 0.5f;
      x = fminf(fmaxf(x, 0.0f), (float)(Wc - 1));
      float yn = 2.0f * ((float)h + 0.5f + offy) * (1.0f / (float)Hc) - 1.0f;
      float y  = (yn + 1.0f) * (0.5f * (float)Hc) - 0.5f;
      y = fminf(fmaxf(y, 0.0f), (float)(Hc - 1));
      const float xf = floorf(x), yf = floorf(y);
      ix0[sy][sx] = (int)xf;
      iy0[sy][sx] = (int)yf;
      ix1[sy][sx] = min(ix0[sy][sx] + 1, Wc - 1);
      iy1[sy][sx] = min(iy0[sy][sx] + 1, Hc - 1);
      fx[sy][sx]  = x - xf;
      fy[sy][sx]  = y - yf;
    }
  }

  // Half-pixel x2 upsample taps: row oy=2h -> (h-1:0.25, h:0.75); oy=2h+1 -> (h:0.75, h+1:0.25)
  const int rym = max(h - 1, 0) - r0;
  const int ry0 = h - r0;
  const int ryp = min(h + 1, Hc - 1) - r0;
  const int cxm = max(w - 1, 0) - c0;
  const int cx0 = w - c0;
  const int cxp = min(w + 1, Wc - 1) - c0;

  int buf = 0;
  for (int ci = 0; ci < CPG; ++ci) {
    if (wave0 && (ci + 1) < CPG) {   // prefetch next channel's hf tile via TDM
      const float* hf_g = high_freq +
          (((size_t)(b * Cc + g * CPG + ci + 1) * 3) * Hc + r0) * (size_t)Wc + c0;
      tdm_load_tile(buf ? lds_hf_addr0 : lds_hf_addr1, hf_g,
                    (u32)rw, (u32)rh, 3, 0,
                    (u64)Wc, (u64)Hc * Wc, 0,
                    Wc, Hc, 3, 0);
    }

    const int c = g * CPG + ci;
    const float* fpl = feature + (size_t)(b * Cc + c) * (Hc * Wc);
    if (ci + 1 < CPG)  // warm the next feature plane (L2 speculative prefetch)
      __builtin_prefetch(fpl + Hc * Wc + (h * Wc + w), 0, 0);

    // Irregular bilinear gathers from the feature plane (L2-resident)
    float res[2][2];
#pragma unroll
    for (int sy = 0; sy < 2; ++sy) {
#pragma unroll
      for (int sx = 0; sx < 2; ++sx) {
        const float v00 = fpl[iy0[sy][sx] * Wc + ix0[sy][sx]];
        const float v01 = fpl[iy0[sy][sx] * Wc + ix1[sy][sx]];
        const float v10 = fpl[iy1[sy][sx] * Wc + ix0[sy][sx]];
        const float v11 = fpl[iy1[sy][sx] * Wc + ix1[sy][sx]];
        const float wxv = fx[sy][sx], wyv = fy[sy][sx];
        const float top = v00 + wxv * (v01 - v00);
        const float bot = v10 + wxv * (v11 - v10);
        res[sy][sx] = top + wyv * (bot - top);
      }
    }

    // High-freq: sum of 3 bands, separable 0.25/0.75 upsample from LDS tile
    float h00 = 0.f, h01 = 0.f, h10 = 0.f, h11 = 0.f;
    const float* T = &lds_hf[buf][0];
#pragma unroll
    for (int band = 0; band < 3; ++band) {
      const float* Tb = T + band * rh * rw;
      const float amm = Tb[rym * rw + cxm], am0 = Tb[rym * rw + cx0], amp = Tb[rym * rw + cxp];
      const float a0m = Tb[ry0 * rw + cxm], a00 = Tb[ry0 * rw + cx0], a0p = Tb[ry0 * rw + cxp];
      const float apm = Tb[ryp * rw + cxm], ap0 = Tb[ryp * rw + cx0], app = Tb[ryp * rw + cxp];
      const float rb0m = 0.25f * amm + 0.75f * a0m;
      const float rb00 = 0.25f * am0 + 0.75f * a00;
      const float rb0p = 0.25f * amp + 0.75f * a0p;
      const float rb1m = 0.75f * a0m + 0.25f * apm;
      const float rb10 = 0.75f * a00 + 0.25f * ap0;
      const float rb1p = 0.75f * a0p + 0.25f * app;
      h00 += 0.25f * rb0m + 0.75f * rb00;
      h01 += 0.75f * rb00 + 0.25f * rb0p;
      h10 += 0.25f * rb1m + 0.75f * rb10;
      h11 += 0.75f * rb10 + 0.25f * rb1p;
    }

    const float kf = 0.1f / 3.0f;   // 0.1 * mean over 3 bands
    float* op = out + (size_t)(b * Cc + c) * (sHc * sWc);
    const int oy0 = 2 * h, ox0 = 2 * w;
    float2 r0v = make_float2(res[0][0] + kf * h00, res[0][1] + kf * h01);
    float2 r1v = make_float2(res[1][0] + kf * h10, res[1][1] + kf * h11);
    *(float2*)(op + (size_t)oy0 * sWc + ox0)       = r0v;
    *(float2*)(op + (size_t)(oy0 + 1) * sWc + ox0) = r1v;

    if (wave0) __builtin_amdgcn_s_wait_tensorcnt(0);
    __syncthreads();
    buf ^= 1;
  }
}

extern "C" void kernel_launch(void* const* d_in, const int* in_sizes, int n_in,
                              void* d_out, int out_size, void* d_ws, size_t ws_size,
                              hipStream_t stream) {
  const float* feature   = (const float*)d_in[0];  // [4,64,160,160]
  const float* offset    = (const float*)d_in[1];  // [4,32,160,160]
  // d_in[2] = low_freq: unused by the reference
  const float* high_freq = (const float*)d_in[3];  // [4,64,3,160,160]
  float* out = (float*)d_out;                      // [4,64,320,320] f32

  dim3 grid((Hc / TS) * (Wc / TS), Gc, Bc);        // 100 x 4 x 4 blocks
  dim3 block(256);
  freq_resampler_kernel<<<grid, block, 0, stream>>>(feature, offset, high_freq, out);
}